// GCN_59545426592262
// MI455X (gfx1250) — compile-verified
//
#include <hip/hip_runtime.h>

// CDNA5 / gfx1250, wave32.
typedef __attribute__((ext_vector_type(2))) float v2f;
typedef __attribute__((ext_vector_type(8))) float v8f;

#define D 128

// ---------------------------------------------------------------------------
// Kernel 1: zero the aggregation buffer (d_ws) with wide stores.
// ---------------------------------------------------------------------------
__global__ void GCN_zero_kernel(float4* __restrict__ p, int n4) {
    int i = blockIdx.x * blockDim.x + threadIdx.x;
    if (i < n4) p[i] = make_float4(0.f, 0.f, 0.f, 0.f);
}

// ---------------------------------------------------------------------------
// Kernel 2: edge scatter-add. One wave32 per edge: each lane reads a float4
// (16B) of x[src] (coalesced 512B per wave) and issues 4 hardware f32 atomics
// into agg[dst]. agg (25.6 MB) is L2-resident on MI455X (192 MB L2), so the
// atomics resolve in L2 without HBM round trips.
// ---------------------------------------------------------------------------
__global__ __launch_bounds__(256) void GCN_scatter_kernel(
    const float* __restrict__ x,
    const int* __restrict__ src,
    const int* __restrict__ dst,
    float* __restrict__ agg,
    int n_edges)
{
    const int gwave = (blockIdx.x * blockDim.x + threadIdx.x) >> 5;
    const int lane  = threadIdx.x & 31;
    if (gwave >= n_edges) return;

    const int s = src[gwave];   // broadcast load (all lanes same address)
    const int d = dst[gwave];

    const float4 v = *((const float4*)(x + (size_t)s * D) + lane);
    float* a = agg + (size_t)d * D + lane * 4;
    unsafeAtomicAdd(a + 0, v.x);   // global_atomic_add_f32
    unsafeAtomicAdd(a + 1, v.y);
    unsafeAtomicAdd(a + 2, v.z);
    unsafeAtomicAdd(a + 3, v.w);
}

// ---------------------------------------------------------------------------
// Kernel 3: fused h = relu(agg @ W^T) + x using V_WMMA_F32_16X16X4_F32.
//
// Block = 256 threads = 8 wave32s. Block b computes output rows
// [16b, 16b+16); wave w computes output cols [16w, 16w+16). K = 128 is
// consumed 4 at a time -> 32 chained WMMAs per wave (f32 in, f32 out, exact
// precision vs. the f32 reference up to summation order).
//
// ISA fragment layouts (CDNA5 ISA §7.12.2):
//   A (16x4 f32):  lanes 0-15: M=lane,  VGPR0=K0, VGPR1=K1
//                  lanes 16-31: M=lane-16, VGPR0=K2, VGPR1=K3
//   B (4x16 f32):  N = lane%16, VGPR0 = K(2*(lane/16)), VGPR1 = K(+1)
//   => both fragments load float2 at row*(D) + k + 2*(lane/16), where the
//      "row" is an agg row for A and a W row (out-dim major) for B, because
//      h[m,n] = sum_k agg[m,k] * W[n,k].
//   C/D: VGPR v holds (row = m0 + v + 8*(lane/16), col = n0 + lane%16).
//
// No divergent control flow -> EXEC all-1s at every WMMA (ISA requirement).
// Requires n_nodes % 16 == 0 (50000 = 3125*16).
// ---------------------------------------------------------------------------
__global__ __launch_bounds__(256) void GCN_gemm_relu_res_kernel(
    const float* __restrict__ agg,
    const float* __restrict__ W,
    const float* __restrict__ x,
    float* __restrict__ out)
{
    const int wave = threadIdx.x >> 5;      // 0..7 -> N tile
    const int lane = threadIdx.x & 31;
    const int m0   = blockIdx.x * 16;
    const int n0   = wave * 16;

    const int r  = lane & 15;               // fragment row (M for A, N for B)
    const int kq = (lane >> 4) << 1;        // K sub-offset: 0 or 2

    const float* aptr = agg + (size_t)(m0 + r) * D + kq;
    const float* bptr = W   + (size_t)(n0 + r) * D + kq;

    v8f c = {};
#pragma unroll
    for (int k = 0; k < D; k += 4) {
        v2f a = *(const v2f*)(aptr + k);
        v2f b = *(const v2f*)(bptr + k);
        // (neg_a, A, neg_b, B, c_mod, C, reuse_a, reuse_b)
        c = __builtin_amdgcn_wmma_f32_16x16x4_f32(
                false, a, false, b, (short)0, c, false, false);
    }

    // Epilogue: ReLU + residual. Lanes 0-15 cover rows m0+v, lanes 16-31
    // rows m0+8+v; each wave's 16 columns are contiguous (coalesced 64B).
    const int rowb = m0 + ((lane >> 4) << 3);
    const int col  = n0 + (lane & 15);
#pragma unroll
    for (int v = 0; v < 8; ++v) {
        const size_t idx = (size_t)(rowb + v) * D + col;
        float h = c[v];
        h = h > 0.f ? h : 0.f;
        out[idx] = h + x[idx];
    }
}

// ---------------------------------------------------------------------------
// Launcher. Inputs (setup_inputs order): x [N*D] f32, src [E] int, dst [E]
// int, W [D*D] f32. Output: h [N*D] f32. Scratch: agg [N*D] f32 in d_ws.
// ---------------------------------------------------------------------------
extern "C" void kernel_launch(void* const* d_in, const int* in_sizes, int n_in,
                              void* d_out, int out_size, void* d_ws, size_t ws_size,
                              hipStream_t stream) {
    const float* x   = (const float*)d_in[0];
    const int*   src = (const int*)d_in[1];
    const int*   dst = (const int*)d_in[2];
    const float* W   = (const float*)d_in[3];
    float* out = (float*)d_out;
    float* agg = (float*)d_ws;

    const int n_nodes = in_sizes[0] / D;    // 50000
    const int n_edges = in_sizes[1];        // 800000

    // 1) agg = 0
    const int n4 = n_nodes * (D / 4);
    GCN_zero_kernel<<<(n4 + 255) / 256, 256, 0, stream>>>((float4*)agg, n4);

    // 2) agg[dst] += x[src]   (one wave per edge)
    const long long threads = (long long)n_edges * 32;
    GCN_scatter_kernel<<<(int)((threads + 255) / 256), 256, 0, stream>>>(
        x, src, dst, agg, n_edges);

    // 3) out = relu(agg @ W^T) + x   (one 16-row block per workgroup)
    GCN_gemm_relu_res_kernel<<<n_nodes / 16, 256, 0, stream>>>(agg, W, x, out);
}